// ContrastiveRegressiveModel_88665304859078
// MI455X (gfx1250) — compile-verified
//
#include <hip/hip_runtime.h>
#include <math.h>

typedef __attribute__((ext_vector_type(2))) float v2f;
typedef __attribute__((ext_vector_type(8))) float v8f;

#define MDIM 4
#define PDIM 128
#define CDIM 128
#define NDIM 256  // 2*P

// One block per (m, i) pair: p in [0,8), m = p>>1, i = (p&1)*P.
// Computes partial[p] = sum_{l != i} ( -L1(ref, batch[m,l]) - log(sum3[l]) )
__global__ __launch_bounds__(256) void crm_pair_kernel(
    const float* __restrict__ q, const float* __restrict__ k,
    const float* __restrict__ labels, float* __restrict__ partial) {
  __shared__ float s_ref[CDIM];
  __shared__ float s_sim[NDIM];   // exp(-L1), 0 on diagonal j==i
  __shared__ float s_log[NDIM];   // -L1  (== log(sims) off-diagonal)
  __shared__ float s_d[NDIM];     // |lab[m,i] - lab[m,j]|
  __shared__ float s_sum3[NDIM];
  __shared__ float s_acc;

  const int p     = blockIdx.x;
  const int m     = p >> 1;
  const int half  = p & 1;          // 0 -> i = 0 (row 0 of q), 1 -> i = P (row 0 of k)
  const int i_idx = half * PDIM;
  const int t     = threadIdx.x;    // 0..255

  // Stage reference row batch[m, i, :] into LDS.
  const float* refp = (half ? k : q) + (size_t)m * PDIM * CDIM;
  if (t < CDIM) s_ref[t] = refp[t];
  if (t == 0) s_acc = 0.f;
  __syncthreads();

  // Thread j computes L1 distance row, sims, and label distances.
  {
    const int j = t;
    const float* bj = (j < PDIM)
                          ? q + ((size_t)m * PDIM + j) * CDIM
                          : k + ((size_t)m * PDIM + (j - PDIM)) * CDIM;
    float l1 = 0.f;
#pragma unroll 8
    for (int c = 0; c < CDIM; ++c) l1 += fabsf(s_ref[c] - bj[c]);
    s_log[j] = -l1;
    s_sim[j] = (j == i_idx) ? 0.f : expf(-l1);
    // labels layout: labels[a, m, pp] at a*(M*P) + m*P + pp
    const float labi = labels[half * (MDIM * PDIM) + m * PDIM];
    const float labj =
        labels[(j >> 7) * (MDIM * PDIM) + m * PDIM + (j & (PDIM - 1))];
    s_d[j] = fabsf(labi - labj);
  }
  __syncthreads();

  // sum3 = PT^T * sims via V_WMMA_F32_16X16X4_F32.
  // A[l_r, kk] = (d[j0+kk] >= d[l0+l_r]) ? 1 : 0   (16x4 comparison tile)
  // B[kk, n]   = sims[j0+kk]  (replicated across the 16 N columns)
  // A layout: lanes 0-15 hold M rows, VGPR {0,1} = K {0,1}; lanes 16-31 = K {2,3}.
  const int wave  = t >> 5;
  const int lane  = t & 31;
  const int lhalf = lane >> 4;  // 0: lanes 0-15, 1: lanes 16-31
  const int lrow  = lane & 15;  // matrix row (A) / N column (D)

  for (int tile = wave * 2; tile < wave * 2 + 2; ++tile) {
    const int l0   = tile * 16;
    const float dl = s_d[l0 + lrow];
    v8f acc = {0.f, 0.f, 0.f, 0.f, 0.f, 0.f, 0.f, 0.f};
    for (int j0 = 0; j0 < NDIM; j0 += 4) {
      const int jb = j0 + lhalf * 2;
      v2f a, b;
      a.x = (s_d[jb + 0] >= dl) ? 1.f : 0.f;
      a.y = (s_d[jb + 1] >= dl) ? 1.f : 0.f;
      b.x = s_sim[jb + 0];
      b.y = s_sim[jb + 1];
      // 8 args: (neg_a, A, neg_b, B, c_mod, C, reuse_a, reuse_b)
      acc = __builtin_amdgcn_wmma_f32_16x16x4_f32(
          false, a, false, b, (short)0, acc, false, false);
    }
    // D rows replicated across columns; take N=0: lane 0 holds M=0..7,
    // lane 16 holds M=8..15, in acc[0..7].
    if (lrow == 0) {
#pragma unroll
      for (int v = 0; v < 8; ++v) s_sum3[l0 + lhalf * 8 + v] = acc[v];
    }
  }
  __syncthreads();

  // Per-l contribution; l == i contributes exactly 0 in the reference.
  float contrib = 0.f;
  if (t != i_idx) contrib = s_log[t] - logf(s_sum3[t]);
  atomicAdd(&s_acc, contrib);
  __syncthreads();
  if (t == 0) partial[p] = s_acc;
}

__global__ void crm_finalize_kernel(const float* __restrict__ partial,
                                    float* __restrict__ out) {
  if (threadIdx.x == 0) {
    float s = 0.f;
#pragma unroll
    for (int p = 0; p < 8; ++p) s += partial[p];
    // loss = scale/2 * total,  scale = -1/(M*(P-1))
    out[0] = s * (-1.0f / (2.0f * MDIM * (PDIM - 1)));
  }
}

extern "C" void kernel_launch(void* const* d_in, const int* in_sizes, int n_in,
                              void* d_out, int out_size, void* d_ws,
                              size_t ws_size, hipStream_t stream) {
  const float* q      = (const float*)d_in[0];
  const float* k      = (const float*)d_in[1];
  const float* labels = (const float*)d_in[2];
  float* out          = (float*)d_out;
  float* partial      = (float*)d_ws;  // 8 floats of scratch

  crm_pair_kernel<<<8, 256, 0, stream>>>(q, k, labels, partial);
  crm_finalize_kernel<<<1, 32, 0, stream>>>(partial, out);
}